// text_graph_61959198212219
// MI455X (gfx1250) — compile-verified
//
#include <hip/hip_runtime.h>

typedef float v2f __attribute__((ext_vector_type(2)));
typedef float v8f __attribute__((ext_vector_type(8)));

#define DIM     512
#define NNODES  32768
#define NEDGES  262144
#define TM      256     // block rows (8 waves x 32-row wave tile)
#define TN      64      // block cols
#define KC      32      // K chunk staged in LDS
#define AS      36      // padded LDS row stride for A tile (floats, 16B aligned)
#define BSP     160     // padded LDS row stride for packed B tile (floats)

// ---------------------------------------------------------------------------
// Degree build: deg[i] = 1 (self loop) + #incoming edges
// ---------------------------------------------------------------------------
__global__ __launch_bounds__(256) void deg_init_kernel(float* __restrict__ deg) {
    int i = blockIdx.x * 256 + threadIdx.x;
    if (i < NNODES) deg[i] = 1.0f;
}

__global__ __launch_bounds__(256) void deg_scatter_kernel(const int* __restrict__ dst,
                                                          float* __restrict__ deg) {
    int e = blockIdx.x * 256 + threadIdx.x;
    if (e < NEDGES) atomicAdd(&deg[dst[e]], 1.0f);
}

// ---------------------------------------------------------------------------
// Tiled f32 WMMA GEMM, 256x64 block tile, 8 waves, 32x64 wave tile.
// B tile is K-pair packed in LDS so each B fragment is a single ds_load_b64
// landing directly in an even-aligned VGPR pair (no repack movs).
// MODE 0: Out0 = prelu(bn(X@W + b1))            (node MLP)
// MODE 1: m = (X@W) * rsqrt(deg[row]); Out0 = m (msg), Out1 = m (agg self-loop)
// ---------------------------------------------------------------------------
template <int MODE>
__global__ __launch_bounds__(256) void gemm_wmma_kernel(
    const float* __restrict__ Xin, const float* __restrict__ Wmat,
    float* __restrict__ Out0, float* __restrict__ Out1,
    const float* __restrict__ p0,   // MODE0: b1        MODE1: deg
    const float* __restrict__ p1,   // MODE0: gamma
    const float* __restrict__ p2,   // MODE0: beta
    const float* __restrict__ p3,   // MODE0: run_mean
    const float* __restrict__ p4,   // MODE0: run_var
    const float* __restrict__ alpha)// MODE0: alpha1    MODE1: unused
{
    __shared__ float As_[TM * AS];              // 36 KB
    __shared__ float Bp_[(KC / 2) * BSP];       // 10 KB, K-pair packed

    const int tid  = threadIdx.x;
    const int wave = tid >> 5;
    const int lane = tid & 31;
    const int col  = lane & 15;   // N / M position within fragment
    const int kh   = lane >> 4;   // K-pair half (0: K{0,1}, 1: K{2,3})
    const int rowBase = blockIdx.y * TM;
    const int colBase = blockIdx.x * TN;

    v8f acc[2][4] = {};

    for (int k0 = 0; k0 < DIM; k0 += KC) {
        __syncthreads();
        // ---- cooperative load A tile: 256 x 32, coalesced float4 ----
        {
            const int r = tid >> 3;          // 0..31
            const int c = (tid & 7) * 4;     // 0..28
            #pragma unroll
            for (int rr = 0; rr < TM; rr += 32) {
                const float4 v = *reinterpret_cast<const float4*>(
                    &Xin[(size_t)(rowBase + r + rr) * DIM + k0 + c]);
                float* d = &As_[(r + rr) * AS + c];
                d[0] = v.x; d[1] = v.y; d[2] = v.z; d[3] = v.w;
                if (k0 + KC < DIM)   // prefetch next K chunk (global_prefetch_b8)
                    __builtin_prefetch(&Xin[(size_t)(rowBase + r + rr) * DIM + k0 + KC + c], 0, 3);
            }
        }
        // ---- cooperative load B tile: 32 x 64 -> K-pair packed LDS ----
        {
            const int r = tid >> 4;          // k row 0..15
            const int c = (tid & 15) * 4;    // 0..60
            #pragma unroll
            for (int rr = 0; rr < KC; rr += 16) {
                const int k = r + rr;
                const float4 v = *reinterpret_cast<const float4*>(
                    &Wmat[(size_t)(k0 + k) * DIM + colBase + c]);
                float* d = &Bp_[(k >> 1) * BSP + c * 2 + (k & 1)];
                d[0] = v.x; d[2] = v.y; d[4] = v.z; d[6] = v.w;
                if (k0 + KC < DIM)
                    __builtin_prefetch(&Wmat[(size_t)(k0 + KC + k) * DIM + colBase + c], 0, 3);
            }
        }
        __syncthreads();

        // ---- 8 K-steps of V_WMMA_F32_16X16X4_F32 per chunk ----
        #pragma unroll
        for (int kk = 0; kk < KC; kk += 4) {
            v2f a0 = *reinterpret_cast<const v2f*>(
                &As_[(wave * 32 + col) * AS + kk + 2 * kh]);
            v2f a1 = *reinterpret_cast<const v2f*>(
                &As_[(wave * 32 + 16 + col) * AS + kk + 2 * kh]);
            #pragma unroll
            for (int n = 0; n < 4; ++n) {
                const v2f b = *reinterpret_cast<const v2f*>(
                    &Bp_[((kk >> 1) + kh) * BSP + (n * 16 + col) * 2]);
                acc[0][n] = __builtin_amdgcn_wmma_f32_16x16x4_f32(
                    false, a0, false, b, (short)0, acc[0][n], false, false);
                acc[1][n] = __builtin_amdgcn_wmma_f32_16x16x4_f32(
                    false, a1, false, b, (short)0, acc[1][n], false, false);
            }
        }
    }

    // ---- fused epilogue ----
    const float a1v = alpha[0];
    #pragma unroll
    for (int n = 0; n < 4; ++n) {
        const int c = colBase + n * 16 + col;
        float mul = 1.0f, add = 0.0f;
        if (MODE == 0) {
            const float sc = p1[c] * rsqrtf(p4[c] + 1e-5f);     // gamma * rsqrt(var+eps)
            mul = sc;
            add = p2[c] + (p0[c] - p3[c]) * sc;                 // beta + (b1-mean)*sc
        }
        #pragma unroll
        for (int s = 0; s < 2; ++s) {
            #pragma unroll
            for (int r = 0; r < 8; ++r) {
                const int row = rowBase + wave * 32 + s * 16 + kh * 8 + r;
                float v = acc[s][n][r];
                if (MODE == 0) {
                    v = v * mul + add;
                    v = (v >= 0.0f) ? v : a1v * v;              // PReLU
                    Out0[(size_t)row * DIM + c] = v;
                } else {
                    const float dinv = rsqrtf(p0[row]);
                    const float m = v * dinv;                   // D^{-1/2} msg
                    Out0[(size_t)row * DIM + c] = m;            // msg buffer
                    Out1[(size_t)row * DIM + c] = m;            // agg init = self loop
                }
            }
        }
    }
}

// ---------------------------------------------------------------------------
// Edge scatter: one block per edge, agg[dst] += msg[src]
// ---------------------------------------------------------------------------
__global__ __launch_bounds__(256) void edge_scatter_kernel(
    const int* __restrict__ src, const int* __restrict__ dst,
    const float* __restrict__ msg, float* __restrict__ agg)
{
    const int e = blockIdx.x;
    const int s = src[e];
    const int d = dst[e];
    const float* mp = &msg[(size_t)s * DIM];
    float* ap = &agg[(size_t)d * DIM];
    #pragma unroll
    for (int c = threadIdx.x; c < DIM; c += 256)
        atomicAdd(&ap[c], mp[c]);
}

// ---------------------------------------------------------------------------
// Finalize (in-place on d_out): g = prelu(agg*dinv + bg); L2 norm; + residual
// ---------------------------------------------------------------------------
__global__ __launch_bounds__(256) void finalize_kernel(
    float* __restrict__ agg, const float* __restrict__ deg,
    const float* __restrict__ bg, const float* __restrict__ alpha2,
    const float* __restrict__ text)
{
    const int row = blockIdx.x;
    const float dinv = rsqrtf(deg[row]);
    const float a2 = alpha2[0];

    __shared__ float red[8];
    float vals[2];
    float ss = 0.0f;
    #pragma unroll
    for (int i = 0; i < 2; ++i) {
        const int c = threadIdx.x + i * 256;
        float g = agg[(size_t)row * DIM + c] * dinv + bg[c];
        g = (g >= 0.0f) ? g : a2 * g;
        vals[i] = g;
        ss += g * g;
    }
    #pragma unroll
    for (int off = 16; off > 0; off >>= 1)
        ss += __shfl_down(ss, off, 32);
    if ((threadIdx.x & 31) == 0) red[threadIdx.x >> 5] = ss;
    __syncthreads();
    float tot = 0.0f;
    #pragma unroll
    for (int i = 0; i < 8; ++i) tot += red[i];
    const float inv = 1.0f / fmaxf(sqrtf(tot), 1e-12f);
    __syncthreads();

    #pragma unroll
    for (int i = 0; i < 2; ++i) {
        const int c = threadIdx.x + i * 256;
        agg[(size_t)row * DIM + c] = vals[i] * inv + text[(size_t)row * DIM + c];
    }
}

// ---------------------------------------------------------------------------
extern "C" void kernel_launch(void* const* d_in, const int* in_sizes, int n_in,
                              void* d_out, int out_size, void* d_ws, size_t ws_size,
                              hipStream_t stream) {
    const float* X        = (const float*)d_in[0];
    const int*   edge_src = (const int*)  d_in[1];
    const int*   edge_dst = (const int*)  d_in[2];
    const float* W1       = (const float*)d_in[3];
    const float* b1       = (const float*)d_in[4];
    const float* gamma    = (const float*)d_in[5];
    const float* beta     = (const float*)d_in[6];
    const float* run_mean = (const float*)d_in[7];
    const float* run_var  = (const float*)d_in[8];
    const float* alpha1   = (const float*)d_in[9];
    const float* Wg       = (const float*)d_in[10];
    const float* bg       = (const float*)d_in[11];
    const float* alpha2   = (const float*)d_in[12];

    float* h   = (float*)d_ws;                       // 64 MB
    float* msg = h   + (size_t)NNODES * DIM;         // 64 MB
    float* deg = msg + (size_t)NNODES * DIM;         // 128 KB
    float* agg = (float*)d_out;                      // agg -> final output in place

    deg_init_kernel<<<NNODES / 256, 256, 0, stream>>>(deg);
    deg_scatter_kernel<<<NEDGES / 256, 256, 0, stream>>>(edge_dst, deg);

    dim3 gemm_grid(DIM / TN, NNODES / TM);           // 8 x 128 blocks
    gemm_wmma_kernel<0><<<gemm_grid, 256, 0, stream>>>(
        X, W1, h, nullptr, b1, gamma, beta, run_mean, run_var, alpha1);
    gemm_wmma_kernel<1><<<gemm_grid, 256, 0, stream>>>(
        h, Wg, msg, agg, deg, nullptr, nullptr, nullptr, nullptr, alpha2);

    edge_scatter_kernel<<<NEDGES, 256, 0, stream>>>(edge_src, edge_dst, msg, agg);
    finalize_kernel<<<NNODES, 256, 0, stream>>>(agg, deg, bg, alpha2, X);
}